// SinkhornRouter_2302102471527
// MI455X (gfx1250) — compile-verified
//
#include <hip/hip_runtime.h>
#include <hip/hip_bf16.h>

// ---------------------------------------------------------------------------
// SinkhornRouter for MI455X (gfx1250, wave32, WMMA + TDM)
//   T = 8192 tokens, E = 64 experts, H = 4096 hidden
//   K1: logits = X @ W^T via V_WMMA_F32_16X16X4_F32.
//       W tiles staged into LDS by the Tensor Data Mover (tensor_load_to_lds,
//       double-buffered, s_wait_tensorcnt), fused cost = exp(logits).
//   K2: single-workgroup Sinkhorn loop, everything in LDS/L2
//   K3: per-row top-2 + fp32 softmax + gather
// ---------------------------------------------------------------------------

#define TOKENS 8192
#define NEXP   64
#define HS     4096
#define SK_EPS 1e-8f
#define SK_TOL 1e-4f

#define KC          64              // K-chunk staged per TDM transfer
#define NCHUNK      (HS / KC)       // 64 chunks
#define BROW_STRIDE (KC + 2)        // LDS row stride in floats (TDM pad: +2 DW)

typedef float    v2f __attribute__((ext_vector_type(2)));
typedef float    v8f __attribute__((ext_vector_type(8)));
typedef unsigned v4u __attribute__((ext_vector_type(4)));
typedef unsigned v8u __attribute__((ext_vector_type(8)));

// ---------------------------------------------------------------------------
// Issue one TDM 2D tile load: W[0:64][k0:k0+KC] (row-major, stride HS) into
// LDS at ldsaddr with a 2-DWORD pad every 64 DWORDs (-> 66-float row stride,
// kills the (n*64+k)%64 bank-conflict pattern on the ds_load_b64 reads).
// D# packing per CDNA5 ISA.
// ---------------------------------------------------------------------------
__device__ __forceinline__ void tdm_load_w_tile(const float* W, int k0,
                                                unsigned ldsaddr) {
    unsigned long long ga =
        (unsigned long long)(const void*)W + (unsigned long long)k0 * 4ull;

    v4u g0;
    g0.x = 1u;                                   // count=1 (valid user D#)
    g0.y = ldsaddr;                              // lds_addr [63:32]
    g0.z = (unsigned)ga;                         // global_addr [95:64]
    g0.w = (unsigned)(ga >> 32) | (2u << 30);    // global_addr hi | type=2

    const unsigned d0 = (2u << 16)               // data_size = 4 bytes
                      | (1u << 20)               // pad_enable
                      | (5u << 22)               // pad_interval: 64 DWORDs
                      | (1u << 25);              // pad_amount:   2 DWORDs
    v8u g1 = {
        d0,
        ((unsigned)HS & 0xFFFFu) << 16,          // tensor_dim0[15:0]
        ((unsigned)HS >> 16) | (((unsigned)NEXP & 0xFFFFu) << 16), // dim0 hi | dim1 lo
        ((unsigned)NEXP >> 16) | ((unsigned)KC << 16),             // dim1 hi | tile_dim0
        (unsigned)NEXP,                          // tile_dim1 | tile_dim2=0
        (unsigned)HS,                            // tensor_dim0_stride[31:0]
        0u,                                      // stride0 hi | stride1 lo
        0u                                       // stride1 hi
    };

    // 2D tensor: descriptor groups 2/3 are NULL -> 2-operand form.
    asm volatile("tensor_load_to_lds %0, %1" :: "s"(g0), "s"(g1) : "memory");
}

// ---------------------------------------------------------------------------
// Kernel 1: GEMM logits(T,E) = X(T,H) @ W(E,H)^T with fp32 WMMA.
// 256 threads = 8 waves; each wave owns a 16-row stripe and all 64 output
// columns (4 accumulator tiles), so X (134 MB) is streamed from HBM exactly
// once. W is DMA'd into LDS once per block per K-chunk by the TDM and read
// back with ds_load_b64 (DScnt pipe overlaps the VMEM A-stream).
// Grid = T / (8*16) = 64 blocks.
//
// Fragment layouts per ISA tables:
//   A 16x4 f32 : lanes 0-15 = M, VGPR0/1 = K{k,k+1}; lanes 16-31 = K{k+2,k+3}
//   B 4x16     : mirrored, N = lane&15, B[k][n] = W[n][k]
//   C/D 16x16  : VGPR r -> M = r (lanes 0-15) / r+8 (lanes 16-31), N = lane&15
// ---------------------------------------------------------------------------
__global__ __launch_bounds__(256)
void sr_gemm_logits(const float* __restrict__ X, const float* __restrict__ W,
                    float* __restrict__ logits, float* __restrict__ cost) {
    __shared__ float Bs[2][NEXP * BROW_STRIDE];   // 2 x 16.5 KB of 320 KB LDS

    const int lane = threadIdx.x & 31;
    const int wave = threadIdx.x >> 5;
    const int m    = lane & 15;
    const int half = lane >> 4;

    const int rowTile = blockIdx.x * 8 + wave;    // 0..511
    const int row0    = rowTile * 16;

    const float* __restrict__ arow = X + (size_t)(row0 + m) * HS + 2 * half;

    const unsigned lds0 = (unsigned)(unsigned long long)&Bs[0][0];
    const unsigned lds1 = (unsigned)(unsigned long long)&Bs[1][0];

    v8f acc0 = {}, acc1 = {}, acc2 = {}, acc3 = {};

    if (wave == 0) tdm_load_w_tile(W, 0, lds0);   // prologue: chunk 0 -> buf 0

    for (int c = 0; c < NCHUNK; ++c) {
        if (wave == 0) __builtin_amdgcn_s_wait_tensorcnt(0);  // chunk c landed
        __syncthreads();   // buf[c&1] ready AND everyone done with buf[~c&1]
        if (wave == 0 && (c + 1) < NCHUNK)
            tdm_load_w_tile(W, (c + 1) * KC, (c & 1) ? lds0 : lds1);
        // TDM for chunk c+1 now overlaps the compute on chunk c.

        const float* __restrict__ bs = &Bs[c & 1][0];
        const int k0 = c * KC;
        #pragma unroll
        for (int kk = 0; kk < KC; kk += 4) {
            v2f a  = *(const v2f*)(arow + k0 + kk);
            v2f b0 = *(const v2f*)(bs + ( 0 + m) * BROW_STRIDE + kk + 2 * half);
            v2f b1 = *(const v2f*)(bs + (16 + m) * BROW_STRIDE + kk + 2 * half);
            v2f b2 = *(const v2f*)(bs + (32 + m) * BROW_STRIDE + kk + 2 * half);
            v2f b3 = *(const v2f*)(bs + (48 + m) * BROW_STRIDE + kk + 2 * half);
            // (neg_a, A, neg_b, B, c_mod, C, reuse_a, reuse_b)
            acc0 = __builtin_amdgcn_wmma_f32_16x16x4_f32(false, a, false, b0,
                                                         (short)0, acc0, false, false);
            acc1 = __builtin_amdgcn_wmma_f32_16x16x4_f32(false, a, false, b1,
                                                         (short)0, acc1, false, false);
            acc2 = __builtin_amdgcn_wmma_f32_16x16x4_f32(false, a, false, b2,
                                                         (short)0, acc2, false, false);
            acc3 = __builtin_amdgcn_wmma_f32_16x16x4_f32(false, a, false, b3,
                                                         (short)0, acc3, false, false);
        }
    }

    // Epilogue: scatter per ISA C/D layout, fuse cost = exp(logits).
    const int n = lane & 15;
    #pragma unroll
    for (int r = 0; r < 8; ++r) {
        const int row = row0 + r + 8 * half;
        float v0 = acc0[r], v1 = acc1[r], v2 = acc2[r], v3 = acc3[r];
        size_t base = (size_t)row * NEXP;
        logits[base +  0 + n] = v0;  cost[base +  0 + n] = expf(v0);
        logits[base + 16 + n] = v1;  cost[base + 16 + n] = expf(v1);
        logits[base + 32 + n] = v2;  cost[base + 32 + n] = expf(v2);
        logits[base + 48 + n] = v3;  cost[base + 48 + n] = expf(v3);
    }
}

// ---------------------------------------------------------------------------
// Kernel 2: Sinkhorn loop. Single 256-thread workgroup; cost (2 MB) is
// L2-resident, d0 (32 KB) lives in LDS. Reference semantics: initial d0 is
// dead (first body iteration only uses d1 = ones), loop runs until
// mean|d1_old - d1_new| <= 1e-4 (>= 1 iteration since err0 = 1e9).
// ---------------------------------------------------------------------------
__global__ __launch_bounds__(256)
void sr_sinkhorn(const float* __restrict__ cost,
                 float* __restrict__ d0g, float* __restrict__ d1g) {
    __shared__ float s_d0[TOKENS];        // 32 KB
    __shared__ float s_d1[NEXP];
    __shared__ float s_colpart[4 * NEXP];
    __shared__ float s_errv[NEXP];
    __shared__ float s_err;

    const int t = threadIdx.x;
    if (t < NEXP) s_d1[t] = 1.0f;
    __syncthreads();

    const float inv_n0 = 1.0f / (float)TOKENS;
    const float inv_n1 = 1.0f / (float)NEXP;

    for (int iter = 0; iter < 500; ++iter) {
        // ---- row pass: d0_new[r] = (1/n0) / (sum_j d1[j]*cost[r][j] + eps)
        for (int i = 0; i < TOKENS / 256; ++i) {
            const int r = t + i * 256;
            const float* __restrict__ cr = cost + (size_t)r * NEXP;
            float s = 0.0f;
            #pragma unroll
            for (int j = 0; j < NEXP; ++j) s += s_d1[j] * cr[j];
            s_d0[r] = inv_n0 / (s + SK_EPS);
        }
        __syncthreads();

        // ---- column pass: 4 threads per column, coalesced 256B row reads
        {
            const int j = t & 63;
            const int g = t >> 6;           // 0..3
            float part = 0.0f;
            for (int r = g; r < TOKENS; r += 4)
                part += s_d0[r] * cost[(size_t)r * NEXP + j];
            s_colpart[g * NEXP + j] = part;
        }
        __syncthreads();

        if (t < NEXP) {
            float csum = s_colpart[t] + s_colpart[NEXP + t] +
                         s_colpart[2 * NEXP + t] + s_colpart[3 * NEXP + t];
            float d1n  = inv_n1 / (csum + SK_EPS);
            s_errv[t]  = fabsf(s_d1[t] - d1n);
            s_d1[t]    = d1n;
        }
        __syncthreads();

        if (t == 0) {
            float e = 0.0f;
            #pragma unroll
            for (int q = 0; q < NEXP; ++q) e += s_errv[q];
            s_err = e * inv_n1;
        }
        __syncthreads();

        if (s_err <= SK_TOL) break;       // uniform across the workgroup
        __syncthreads();
    }

    // publish final normalizers
    for (int i = 0; i < TOKENS / 256; ++i) {
        const int r = t + i * 256;
        d0g[r] = s_d0[r];
    }
    if (t < NEXP) d1g[t] = s_d1[t];
}

// ---------------------------------------------------------------------------
// Kernel 3: per-row top-2 of d1[j]*cost[i][j]*d0[i] (strict > ascending scan
// reproduces jax.lax.top_k lowest-index tie-break), fp32 softmax of logits,
// gather scores at the two indices.
// ---------------------------------------------------------------------------
__global__ __launch_bounds__(256)
void sr_finalize(const float* __restrict__ logits, const float* __restrict__ cost,
                 const float* __restrict__ d0g, const float* __restrict__ d1g,
                 float* __restrict__ scores, int* __restrict__ indices) {
    __shared__ float s_d1[NEXP];
    if (threadIdx.x < NEXP) s_d1[threadIdx.x] = d1g[threadIdx.x];
    __syncthreads();

    const int i = blockIdx.x * blockDim.x + threadIdx.x;   // 0..8191
    const float* __restrict__ lr = logits + (size_t)i * NEXP;
    const float* __restrict__ cr = cost   + (size_t)i * NEXP;
    const float d0 = d0g[i];

    float b1 = -3.402823466e38f, b2 = -3.402823466e38f;
    int   i1 = 0, i2 = 0;
    #pragma unroll
    for (int j = 0; j < NEXP; ++j) {
        const float v = s_d1[j] * cr[j] * d0;
        if (v > b1)      { b2 = b1; i2 = i1; b1 = v; i1 = j; }
        else if (v > b2) { b2 = v;  i2 = j; }
    }

    float mx = -3.402823466e38f;
    #pragma unroll
    for (int j = 0; j < NEXP; ++j) mx = fmaxf(mx, lr[j]);
    float denom = 0.0f;
    #pragma unroll
    for (int j = 0; j < NEXP; ++j) denom += expf(lr[j] - mx);
    const float inv = 1.0f / denom;

    scores[2 * i + 0]  = expf(lr[i1] - mx) * inv;
    scores[2 * i + 1]  = expf(lr[i2] - mx) * inv;
    indices[2 * i + 0] = i1;
    indices[2 * i + 1] = i2;
}

// ---------------------------------------------------------------------------
extern "C" void kernel_launch(void* const* d_in, const int* in_sizes, int n_in,
                              void* d_out, int out_size, void* d_ws, size_t ws_size,
                              hipStream_t stream) {
    (void)in_sizes; (void)n_in; (void)out_size; (void)ws_size;

    const float* X = (const float*)d_in[0];   // (8192, 4096) fp32 row-major
    const float* W = (const float*)d_in[1];   // (64, 4096)   fp32 row-major

    float* ws     = (float*)d_ws;
    float* logits = ws;                                   // T*E floats (2 MB)
    float* cost   = logits + (size_t)TOKENS * NEXP;       // T*E floats (2 MB)
    float* d0g    = cost   + (size_t)TOKENS * NEXP;       // T floats
    float* d1g    = d0g    + TOKENS;                      // E floats

    float* scores  = (float*)d_out;                       // (T,2) fp32
    int*   indices = ((int*)d_out) + 2 * TOKENS;          // (T,2) int32

    sr_gemm_logits<<<TOKENS / 128, 256, 0, stream>>>(X, W, logits, cost);
    sr_sinkhorn<<<1, 256, 0, stream>>>(cost, d0g, d1g);
    sr_finalize<<<TOKENS / 256, 256, 0, stream>>>(logits, cost, d0g, d1g,
                                                  scores, indices);
}